// NCC2D_44856638439947
// MI455X (gfx1250) — compile-verified
//
#include <hip/hip_runtime.h>

// ---------------------------------------------------------------------------
// NCC 9x9 loss for [32,1,512,512] fp32 on gfx1250 (MI455X).
// Box filters are banded-matrix multiplies on V_WMMA_F32_16X16X4_F32 (full
// fp32 precision). Band operands hoisted to registers; vertical pass is
// chunk-major across the 5 fields so each LDS read feeds 5 WMMAs; interior
// tiles use 16B-aligned b128 global loads. Deterministic 2-stage reduction.
// ---------------------------------------------------------------------------

typedef __attribute__((ext_vector_type(2))) float v2f;
typedef __attribute__((ext_vector_type(8))) float v8f;

enum : int {
  BATCH = 32,
  HH    = 512,
  WW    = 512,
  TILE  = 16,
  HALO  = 4,                 // WIN/2 with WIN=9
  IN    = TILE + 2 * HALO,   // 24
  LDSW  = 40                 // LDS row stride (floats): 160B rows, 16B aligned
};

__device__ __forceinline__ v8f wmma_f32x4(v2f a, v2f b, v8f c) {
  // D = A(16x4,f32) * B(4x16,f32) + C(16x16,f32)
  return __builtin_amdgcn_wmma_f32_16x16x4_f32(
      /*neg_a=*/false, a, /*neg_b=*/false, b,
      /*c_mod=*/(short)0, c, /*reuse_a=*/false, /*reuse_b=*/false);
}

template <bool ATOMIC>
__global__ __launch_bounds__(32) void ncc_tile_kernel(
    const float* __restrict__ Iimg, const float* __restrict__ Jimg,
    float* __restrict__ partial, float* __restrict__ out) {
  __shared__ float sI[IN][LDSW];
  __shared__ float sJ[IN][LDSW];
  __shared__ float sV[5][TILE][LDSW];  // vertical results, all 5 fields

  const int lane = threadIdx.x;        // wave32: 0..31
  const int mn   = lane & 15;          // M for A/D rows, N for B/D cols
  const int hsel = (lane >> 4) << 1;   // K sub-select per half-wave: 0 or 2
  const int drow = (lane >> 4) << 3;   // D-layout row offset: 0 or 8

  const int r0 = blockIdx.y * TILE - HALO;  // top-left of 24x24 input patch
  const int c0 = blockIdx.x * TILE - HALO;
  const float* Ib = Iimg + (size_t)blockIdx.z * (HH * WW);
  const float* Jb = Jimg + (size_t)blockIdx.z * (HH * WW);

  // ---- stage the 24x24 patch (zero padding = the conv's zero pad) ---------
  const bool interior =
      (r0 >= 0) && (c0 >= 0) && (r0 + IN <= HH) && (c0 + IN <= WW);
  if (interior) {
    // Only pad cols 24..31 are ever read (second N-block); zero just those.
    for (int t = lane; t < IN * 8; t += 32) {
      const int rr = t >> 3, cc = 24 + (t & 7);
      sI[rr][cc] = 0.0f;
      sJ[rr][cc] = 0.0f;
    }
    // 16B-aligned b128 loads: c0 % 4 == 0, row stride 512, LDS rows 160B.
    for (int t = lane; t < IN * 6; t += 32) {       // 24 rows x 6 float4
      const int rr = t / 6, q = (t - rr * 6) << 2;  // q = 0,4,...,20
      const size_t off = (size_t)(r0 + rr) * WW + (c0 + q);
      *(float4*)&sI[rr][q] = *(const float4*)(Ib + off);
      *(float4*)&sJ[rr][q] = *(const float4*)(Jb + off);
    }
  } else {
    for (int t = lane; t < IN * 32; t += 32) {  // zero cols 0..31
      const int rr = t >> 5, cc = t & 31;
      sI[rr][cc] = 0.0f;
      sJ[rr][cc] = 0.0f;
    }
    __syncthreads();
    for (int t = lane; t < IN * IN; t += 32) {
      const int rr = t / IN, cc = t - rr * IN;
      const int gr = r0 + rr, gc = c0 + cc;
      if (gr >= 0 && gr < HH && gc >= 0 && gc < WW) {
        const size_t off = (size_t)gr * WW + gc;
        sI[rr][cc] = Ib[off];
        sJ[rr][cc] = Jb[off];
      }
    }
  }
  __syncthreads();

  // ---- hoisted band operand: Band[m][k] = (m <= k <= m+8) -----------------
  // Per lane this is A(M=mn,K) for the vertical pass AND B(K,N=mn) for the
  // horizontal pass (same predicate), for K = 4*kc + hsel + {0,1}.
  v2f band[6];
#pragma unroll
  for (int kc = 0; kc < 6; ++kc) {
    const int k0 = (kc << 2) + hsel;
    band[kc].x = (k0     >= mn && k0     <= mn + 8) ? 1.0f : 0.0f;
    band[kc].y = (k0 + 1 >= mn && k0 + 1 <= mn + 8) ? 1.0f : 0.0f;
  }

  // ---- vertical pass: V_f(16x24) = Band(16x24) x F_f(24x[16|16]) ----------
  // Chunk-major over fields: one pair of LDS reads feeds all 5 WMMAs.
#pragma unroll
  for (int cb = 0; cb < 2; ++cb) {
    const int col = (cb << 4) + mn;
    v8f aI = {}, aJ = {}, aII = {}, aJJ = {}, aIJ = {};
#pragma unroll
    for (int kc = 0; kc < 6; ++kc) {
      const int k0 = (kc << 2) + hsel;
      const float i0 = sI[k0][col], i1 = sI[k0 + 1][col];
      const float j0 = sJ[k0][col], j1 = sJ[k0 + 1][col];
      const v2f bI  = {i0,      i1};
      const v2f bJ  = {j0,      j1};
      const v2f bII = {i0 * i0, i1 * i1};
      const v2f bJJ = {j0 * j0, j1 * j1};
      const v2f bIJ = {i0 * j0, i1 * j1};
      aI  = wmma_f32x4(band[kc], bI,  aI);
      aJ  = wmma_f32x4(band[kc], bJ,  aJ);
      aII = wmma_f32x4(band[kc], bII, aII);
      aJJ = wmma_f32x4(band[kc], bJJ, aJJ);
      aIJ = wmma_f32x4(band[kc], bIJ, aIJ);
    }
#pragma unroll
    for (int v = 0; v < 8; ++v) {
      sV[0][v + drow][col] = aI[v];
      sV[1][v + drow][col] = aJ[v];
      sV[2][v + drow][col] = aII[v];
      sV[3][v + drow][col] = aJJ[v];
      sV[4][v + drow][col] = aIJ[v];
    }
  }
  __syncthreads();

  // ---- horizontal pass: S_f(16x16) = V_f(16x24) x Band^T(24x16) -----------
  v8f S[5] = {{}, {}, {}, {}, {}};
#pragma unroll
  for (int kc = 0; kc < 6; ++kc) {
    const int k0 = (kc << 2) + hsel;
#pragma unroll
    for (int f = 0; f < 5; ++f) {
      v2f a;
      a.x = sV[f][mn][k0];
      a.y = sV[f][mn][k0 + 1];
      S[f] = wmma_f32x4(a, band[kc], S[f]);
    }
  }

  // ---- per-pixel combine (VALU), reference-literal algebra ----------------
  const float win = 81.0f, inv_win = 1.0f / 81.0f;
  float csum = 0.0f;
#pragma unroll
  for (int v = 0; v < 8; ++v) {
    const float sIv = S[0][v], sJv = S[1][v];
    const float sII = S[2][v], sJJ = S[3][v], sIJ = S[4][v];
    const float uI = sIv * inv_win;
    const float uJ = sJv * inv_win;
    const float cross = sIJ - uJ * sIv - uI * sJv + uI * uJ * win;
    const float Ivar  = sII - 2.0f * uI * sIv + uI * uI * win;
    const float Jvar  = sJJ - 2.0f * uJ * sJv + uJ * uJ * win;
    csum += cross * cross / (Ivar * Jvar + 1e-5f);
  }
  // wave32 reduction
#pragma unroll
  for (int off = 16; off > 0; off >>= 1)
    csum += __shfl_xor(csum, off, 32);

  if (lane == 0) {
    if (ATOMIC) {
      atomicAdd(out, csum * (-1.0f / (float)((long long)BATCH * HH * WW)));
    } else {
      const int idx =
          (blockIdx.z * gridDim.y + blockIdx.y) * gridDim.x + blockIdx.x;
      partial[idx] = csum;
    }
  }
}

__global__ __launch_bounds__(256) void ncc_reduce_kernel(
    const float* __restrict__ partial, int n, float* __restrict__ out) {
  __shared__ float sm[256];
  float s = 0.0f;
  for (int i = threadIdx.x; i < n; i += 256) s += partial[i];
  sm[threadIdx.x] = s;
  __syncthreads();
  for (int off = 128; off > 0; off >>= 1) {
    if ((int)threadIdx.x < off) sm[threadIdx.x] += sm[threadIdx.x + off];
    __syncthreads();
  }
  if (threadIdx.x == 0)
    out[0] = -sm[0] / (float)((long long)BATCH * HH * WW);
}

__global__ void ncc_zero_out(float* out) {
  if (threadIdx.x == 0) out[0] = 0.0f;
}

extern "C" void kernel_launch(void* const* d_in, const int* in_sizes, int n_in,
                              void* d_out, int out_size, void* d_ws,
                              size_t ws_size, hipStream_t stream) {
  const float* I = (const float*)d_in[0];  // y_true
  const float* J = (const float*)d_in[1];  // y_pred
  float* out = (float*)d_out;
  float* ws  = (float*)d_ws;

  dim3 grid(WW / TILE, HH / TILE, BATCH);  // 32 x 32 x 32 tiles
  const int nblk = (WW / TILE) * (HH / TILE) * BATCH;

  if (ws_size >= (size_t)nblk * sizeof(float)) {
    ncc_tile_kernel<false><<<grid, 32, 0, stream>>>(I, J, ws, out);
    ncc_reduce_kernel<<<1, 256, 0, stream>>>(ws, nblk, out);
  } else {
    ncc_zero_out<<<1, 32, 0, stream>>>(out);
    ncc_tile_kernel<true><<<grid, 32, 0, stream>>>(I, J, ws, out);
  }
}